// CTC_73710228734773
// MI455X (gfx1250) — compile-verified
//
#include <hip/hip_runtime.h>
#include <math.h>

typedef __attribute__((ext_vector_type(2))) float v2f;
typedef __attribute__((ext_vector_type(8))) float v8f;
typedef int av4i __attribute__((vector_size(16)));                 // int __vector(4)
typedef __attribute__((address_space(1))) av4i* gv4i_ptr;          // global int4*
typedef __attribute__((address_space(3))) av4i* lv4i_ptr;          // LDS int4*

#define BB     16
#define TT     512
#define DD     512
#define VV     4096
#define LL     64
#define SS     129           // 2*L+1
#define SPAD   132           // padded S
#define MM     (BB*TT)       // 8192 rows
#define NCHUNK 8             // V split into 8 chunks of 512
#define CW     512
#define LSTR   36            // LDS row stride in floats (144B, 16B aligned)
#define NEGV   (-1.0e30f)

__device__ __forceinline__ float laddexp(float a, float b) {
    float m = fmaxf(a, b);
    float n = fminf(a, b);
    return m + log1pf(expf(n - m));
}

// ---- CDNA5 async global->LDS copy (ASYNCcnt path), 16 bytes per lane -------
__device__ __forceinline__ void async_copy_b128(const float* gsrc, float* ldst)
{
#if __has_builtin(__builtin_amdgcn_global_load_async_to_lds_b128)
    __builtin_amdgcn_global_load_async_to_lds_b128(
        (gv4i_ptr)(uintptr_t)gsrc,
        (lv4i_ptr)(unsigned)(uintptr_t)ldst,
        0, 0);
#else
    unsigned l = (unsigned)(uintptr_t)ldst;
    asm volatile("global_load_async_to_lds_b128 %0, %1, off"
                 :: "v"(l), "v"(gsrc) : "memory");
#endif
}

__device__ __forceinline__ void wait_async0()
{
#if __has_builtin(__builtin_amdgcn_s_wait_asynccnt)
    __builtin_amdgcn_s_wait_asynccnt(0);
#else
    asm volatile("s_wait_asynccnt 0x0" ::: "memory");
#endif
}

// ---------------------------------------------------------------------------
// K1: fused GEMM (fp32 WMMA) + running logsumexp partials + label-logit gather
// grid: (NCHUNK, MM/64), block: 256 threads (8 waves)
// Double-buffered LDS tiles filled by async-to-LDS copies overlapping WMMAs.
// ---------------------------------------------------------------------------
__global__ __launch_bounds__(256)
void k_gemm_lse_gather(const float* __restrict__ y, const float* __restrict__ W,
                       const float* __restrict__ bias, const int* __restrict__ tgt,
                       float* __restrict__ partMax, float* __restrict__ partSum,
                       float* __restrict__ lpLabel)
{
    const int chunk = blockIdx.x;          // 0..7  (columns chunk*512 .. +511)
    const int m0    = blockIdx.y * 64;     // first row of tile
    const int batch = m0 / TT;             // all 64 rows belong to one batch
    const int tid   = threadIdx.x;
    const int wave  = tid >> 5;
    const int lane  = tid & 31;
    const int lh    = lane >> 4;           // lane half (0/1)
    const int ll    = lane & 15;           // lane low
    const int wm    = wave >> 1;           // 0..3  (16-row strip)
    const int wn    = wave & 1;            // 0..1  (32-col strip)

    __shared__ float As[2][64][LSTR];      // A tiles (rows x K), double buffered
    __shared__ float Bs[2][64][LSTR];      // B tiles (cols x K), double buffered
    __shared__ float Ct[64][65];           // staged 64x64 logits sub-tile
    __shared__ int   extS[SS];             // extended target labels

    for (int s = tid; s < SS; s += 256)
        extS[s] = (s & 1) ? tgt[batch * LL + (s >> 1)] : 0;   // BLANK = 0

    // per-thread stationary copy mapping: two rows of A, two rows of B
    const int cr  = tid >> 3;              // 0..31
    const int cc4 = (tid & 7) * 4;         // 0..28 (float4 column)

    float runMax = NEGV;                   // valid for tid < 64 (one per row)
    float runSum = 0.0f;

    for (int nIter = 0; nIter < 8; ++nIter) {
        const int n0 = chunk * CW + nIter * 64;
        v8f acc0 = {}; v8f acc1 = {};

        // prologue: stage K-block 0 into buffer 0 (4 async b128 per thread)
        async_copy_b128(&y[(size_t)(m0 + cr) * DD + cc4],      &As[0][cr][cc4]);
        async_copy_b128(&y[(size_t)(m0 + cr + 32) * DD + cc4], &As[0][cr + 32][cc4]);
        async_copy_b128(&W[(size_t)(n0 + cr) * DD + cc4],      &Bs[0][cr][cc4]);
        async_copy_b128(&W[(size_t)(n0 + cr + 32) * DD + cc4], &Bs[0][cr + 32][cc4]);

        for (int kbi = 0; kbi < 16; ++kbi) {
            wait_async0();                 // this thread's tile-kbi copies done
            __syncthreads();               // visible to all waves; buffers free
            if (kbi < 15) {                // overlap copy of kbi+1 with compute
                const int kb = (kbi + 1) * 32;
                const int nb = (kbi + 1) & 1;
                async_copy_b128(&y[(size_t)(m0 + cr) * DD + kb + cc4],      &As[nb][cr][cc4]);
                async_copy_b128(&y[(size_t)(m0 + cr + 32) * DD + kb + cc4], &As[nb][cr + 32][cc4]);
                async_copy_b128(&W[(size_t)(n0 + cr) * DD + kb + cc4],      &Bs[nb][cr][cc4]);
                async_copy_b128(&W[(size_t)(n0 + cr + 32) * DD + kb + cc4], &Bs[nb][cr + 32][cc4]);
            }
            const int buf = kbi & 1;
            #pragma unroll
            for (int k0 = 0; k0 < 32; k0 += 4) {
                const int kk = k0 + 2 * lh;
                v2f a  = *(const v2f*)&As[buf][wm * 16 + ll][kk];
                v2f b0 = *(const v2f*)&Bs[buf][wn * 32 + ll][kk];
                v2f b1 = *(const v2f*)&Bs[buf][wn * 32 + 16 + ll][kk];
                acc0 = __builtin_amdgcn_wmma_f32_16x16x4_f32(false, a, false, b0,
                                                             (short)0, acc0, false, false);
                acc1 = __builtin_amdgcn_wmma_f32_16x16x4_f32(false, a, false, b1,
                                                             (short)0, acc1, false, false);
            }
        }
        __syncthreads();
        // stage logits (+bias) to LDS.  C/D layout: vgpr i -> M = i + 8*lh, N = ll
        {
            const float bv0 = bias[n0 + wn * 32 + ll];
            const float bv1 = bias[n0 + wn * 32 + 16 + ll];
            #pragma unroll
            for (int i = 0; i < 8; ++i) {
                Ct[wm * 16 + 8 * lh + i][wn * 32 + ll]      = acc0[i] + bv0;
                Ct[wm * 16 + 8 * lh + i][wn * 32 + 16 + ll] = acc1[i] + bv1;
            }
        }
        __syncthreads();
        // per-row running logsumexp update + gather label logits (thread = row)
        if (tid < 64) {
            float tmx = NEGV;
            for (int c = 0; c < 64; ++c) tmx = fmaxf(tmx, Ct[tid][c]);
            const float nmx = fmaxf(runMax, tmx);
            float s = runSum * expf(runMax - nmx);
            for (int c = 0; c < 64; ++c) s += expf(Ct[tid][c] - nmx);
            runMax = nmx; runSum = s;

            const int m = m0 + tid;
            for (int s2 = 0; s2 < SS; ++s2) {
                const int d = extS[s2] - n0;
                if (d >= 0 && d < 64)
                    lpLabel[(size_t)m * SPAD + s2] = Ct[tid][d];
            }
        }
    }
    if (tid < 64) {
        partMax[(size_t)(m0 + tid) * NCHUNK + chunk] = runMax;
        partSum[(size_t)(m0 + tid) * NCHUNK + chunk] = runSum;
    }
}

// ---------------------------------------------------------------------------
// K2: combine per-chunk (max,sum) partials into lse per row
// ---------------------------------------------------------------------------
__global__ void k_combine_lse(const float* __restrict__ partMax,
                              const float* __restrict__ partSum,
                              float* __restrict__ lse)
{
    const int m = blockIdx.x * 256 + threadIdx.x;
    if (m >= MM) return;
    float mx = NEGV;
    for (int c = 0; c < NCHUNK; ++c) mx = fmaxf(mx, partMax[(size_t)m * NCHUNK + c]);
    float s = 0.0f;
    for (int c = 0; c < NCHUNK; ++c)
        s += partSum[(size_t)m * NCHUNK + c] * expf(partMax[(size_t)m * NCHUNK + c] - mx);
    lse[m] = mx + logf(s);
}

// ---------------------------------------------------------------------------
// K3: CTC forward DP, one block per batch sample. 160 threads (5 waves),
// thread s < 129 owns state s; alpha in LDS; sequential over t.
// ---------------------------------------------------------------------------
__global__ __launch_bounds__(160)
void k_ctc_dp(const float* __restrict__ lpLabel, const float* __restrict__ lse,
              const int* __restrict__ tgt, const int* __restrict__ ylen,
              const int* __restrict__ tlen, float* __restrict__ nll)
{
    const int b = blockIdx.x;
    const int s = threadIdx.x;
    const bool active = (s < SS);
    __shared__ float alpha[SPAD];
    __shared__ float res[2];

    const int len = ylen[b];
    const int tl  = tlen[b];

    bool skip = false;
    if (active && (s & 1)) {
        const int l = s >> 1;
        const int v = tgt[b * LL + l];
        const int prev = (s >= 3) ? tgt[b * LL + l - 1] : -1;
        skip = (v != 0) && (v != prev);
    }

    const float* lpL = lpLabel + (size_t)(b * TT) * SPAD;
    const float* ls  = lse + (size_t)b * TT;

    if (active) alpha[s] = (s <= 1) ? (lpL[s] - ls[0]) : NEGV;
    __syncthreads();
    if (s == 0 && len == 1) {
        res[0] = alpha[2 * tl];
        res[1] = (tl > 0) ? alpha[2 * tl - 1] : NEGV;
    }

    for (int t = 1; t < TT; ++t) {
        float am0 = NEGV, am1 = NEGV, am2 = NEGV, lpv = 0.0f;
        if (active) {
            am0 = alpha[s];
            am1 = (s >= 1) ? alpha[s - 1] : NEGV;
            am2 = (s >= 2) ? alpha[s - 2] : NEGV;
            lpv = lpL[(size_t)t * SPAD + s] - ls[t];
        }
        __syncthreads();
        if (active) {
            float a = laddexp(am0, am1);
            if (skip) a = laddexp(a, am2);
            alpha[s] = a + lpv;
        }
        __syncthreads();
        if (s == 0 && t == len - 1) {
            res[0] = alpha[2 * tl];
            res[1] = (tl > 0) ? alpha[2 * tl - 1] : NEGV;
        }
    }
    __syncthreads();
    if (s == 0) nll[b] = -laddexp(res[0], res[1]);
}

// ---------------------------------------------------------------------------
// K4: mean over batch
// ---------------------------------------------------------------------------
__global__ void k_finalize(const float* __restrict__ nll, float* __restrict__ out)
{
    if (threadIdx.x == 0) {
        float s = 0.0f;
        for (int i = 0; i < BB; ++i) s += nll[i];
        out[0] = s / (float)BB;
    }
}

extern "C" void kernel_launch(void* const* d_in, const int* in_sizes, int n_in,
                              void* d_out, int out_size, void* d_ws, size_t ws_size,
                              hipStream_t stream)
{
    (void)in_sizes; (void)n_in; (void)out_size; (void)ws_size;
    const float* y    = (const float*)d_in[0];
    const int*   ylen = (const int*)d_in[1];
    const int*   tgt  = (const int*)d_in[2];
    const int*   tlen = (const int*)d_in[3];
    const float* W    = (const float*)d_in[4];
    const float* bias = (const float*)d_in[5];
    float* out = (float*)d_out;

    float* ws      = (float*)d_ws;
    float* partMax = ws;                                   // MM*NCHUNK
    float* partSum = partMax + (size_t)MM * NCHUNK;        // MM*NCHUNK
    float* lse     = partSum + (size_t)MM * NCHUNK;        // MM
    float* lpLabel = lse + MM;                             // MM*SPAD
    float* nll     = lpLabel + (size_t)MM * SPAD;          // BB
    // total ~4.9 MB of d_ws

    k_gemm_lse_gather<<<dim3(NCHUNK, MM / 64), 256, 0, stream>>>(
        y, W, bias, tgt, partMax, partSum, lpLabel);
    k_combine_lse<<<MM / 256, 256, 0, stream>>>(partMax, partSum, lse);
    k_ctc_dp<<<BB, 160, 0, stream>>>(lpLabel, lse, tgt, ylen, tlen, nll);
    k_finalize<<<1, 32, 0, stream>>>(nll, out);
}